// losses_refiner_28106265985710
// MI455X (gfx1250) — compile-verified
//
#include <hip/hip_runtime.h>
#include <hip/hip_bf16.h>

#define BB 32
#define NN 1024

typedef __attribute__((ext_vector_type(2))) float v2f;
typedef __attribute__((ext_vector_type(8))) float v8f;

// ---------------------------------------------------------------------------
// Kernel 1: fused pose transforms + workspace init. One thread per point.
//   q = dR*p + dt ; g = Rg*p + tg ; r = Rc*q + tc
// A-side rows stored as [-2rx,-2ry,-2rz, 1], B-side cols as [gx,gy,gz,|g|^2],
// so V_WMMA_F32_16X16X4_F32 yields acc = |g|^2 - 2*pred.gt directly and
// d^2 = |r|^2 + acc is a single add per element. Seeds BOTH min buffers with
// the +inf bit pattern (uint atomicMin is order-correct for floats >= 0).
// ---------------------------------------------------------------------------
__global__ void pose_points_kernel(
    const float* __restrict__ dR,  // [B,3,3]
    const float* __restrict__ dT,  // [B,3]
    const float* __restrict__ Tc,  // [B,3]
    const float* __restrict__ Rc,  // [B,3,3]
    const float* __restrict__ Rg,  // [B,3,3]
    const float* __restrict__ Tg,  // [B,3]
    const float* __restrict__ P,   // [B,N,3]
    float* __restrict__ pred4,     // [B,N,4] = [-2r, 1]
    float* __restrict__ gt4,       // [B,N,4] = [g, |g|^2]
    float* __restrict__ p2,        // [B,N]   = |r|^2
    float* __restrict__ l2,        // [B,N]
    unsigned int* __restrict__ dis1u,  // [B,N] row-min d^2, seeded +inf
    unsigned int* __restrict__ dis2u)  // [B,N] col-min d^2, seeded +inf
{
    const int idx = blockIdx.x * blockDim.x + threadIdx.x;   // [0, B*N)
    const int b   = idx >> 10;                               // N = 1024

    const float* p = P + idx * 3;
    const float x = p[0], y = p[1], z = p[2];

    const float* A  = dR + b * 9;
    const float* t  = dT + b * 3;
    const float* Cc = Rc + b * 9;
    const float* tc = Tc + b * 3;
    const float* G  = Rg + b * 9;
    const float* tg = Tg + b * 3;

    // posed_pred = dR @ p + dt
    const float qx = A[0]*x + A[1]*y + A[2]*z + t[0];
    const float qy = A[3]*x + A[4]*y + A[5]*z + t[1];
    const float qz = A[6]*x + A[7]*y + A[8]*z + t[2];

    // posed_refined = Rc @ q + tc
    const float rx = Cc[0]*qx + Cc[1]*qy + Cc[2]*qz + tc[0];
    const float ry = Cc[3]*qx + Cc[4]*qy + Cc[5]*qz + tc[1];
    const float rz = Cc[6]*qx + Cc[7]*qy + Cc[8]*qz + tc[2];

    // posed_gt = Rg @ p + tg
    const float gx = G[0]*x + G[1]*y + G[2]*z + tg[0];
    const float gy = G[3]*x + G[4]*y + G[5]*z + tg[1];
    const float gz = G[6]*x + G[7]*y + G[8]*z + tg[2];

    float* pr = pred4 + (idx << 2);
    pr[0] = -2.0f * rx; pr[1] = -2.0f * ry; pr[2] = -2.0f * rz; pr[3] = 1.0f;
    float* gr = gt4 + (idx << 2);
    gr[0] = gx; gr[1] = gy; gr[2] = gz; gr[3] = gx*gx + gy*gy + gz*gz;

    p2[idx] = rx*rx + ry*ry + rz*rz;

    const float dx = rx - gx, dy = ry - gy, dz = rz - gz;
    l2[idx] = __builtin_amdgcn_sqrtf(dx*dx + dy*dy + dz*dz);

    dis1u[idx] = 0x7F800000u;   // +inf
    dis2u[idx] = 0x7F800000u;   // +inf
}

// ---------------------------------------------------------------------------
// Kernel 2: Chamfer via V_WMMA_F32_16X16X4_F32 on SQUARED distances
// (min commutes with sqrt for x>=0; sqrt deferred to the finalize kernel).
// One wave32 per (batch, 32-row strip): TWO 16-row A-tiles in registers, two
// back-to-back WMMAs per B-tile load. All mins tracked as uint bit patterns
// (v_min_u32, no NaN canonicalization). Column mins AND row mins are merged
// through fire-and-forget global atomicMin -- no DS ops, no exec branches
// anywhere in this kernel.
// ---------------------------------------------------------------------------
__global__ __launch_bounds__(32) void chamfer_wmma_kernel(
    const float* __restrict__ pred4,     // [-2r, 1]
    const float* __restrict__ gt4,       // [g, |g|^2]
    const float* __restrict__ p2,        // |r|^2
    unsigned int* __restrict__ dis1u,    // [B,N] row mins of d^2 (bit pattern)
    unsigned int* __restrict__ dis2u)    // [B,N] col mins of d^2 (bit pattern)
{
    const int wave = blockIdx.x;         // B*32 waves
    const int b    = wave >> 5;
    const int n0   = (wave & 31) << 5;   // 32 rows per wave
    const int lane = threadIdx.x;        // wave32
    const int half = lane >> 4;          // 0: K=0,1   1: K=2,3
    const int l15  = lane & 15;
    const int k0   = half << 1;

    // Two A-tiles: rows n0..n0+15 and n0+16..n0+31
    const float* ap0 = pred4 + (((b * NN) + n0 + l15) << 2) + k0;
    const float* ap1 = ap0 + (16 << 2);
    v2f a0, a1;
    a0.x = ap0[0]; a0.y = ap0[1];
    a1.x = ap1[0]; a1.y = ap1[1];

    // |r|^2 for the rows each accumulator slot maps to
    float p2r0[8], p2r1[8];
    #pragma unroll
    for (int r = 0; r < 8; ++r) {
        p2r0[r] = p2[b * NN + n0      + r + half * 8];
        p2r1[r] = p2[b * NN + n0 + 16 + r + half * 8];
    }

    unsigned int rmin0[8], rmin1[8];
    #pragma unroll
    for (int r = 0; r < 8; ++r) { rmin0[r] = 0x7F800000u; rmin1[r] = 0x7F800000u; }

    const float* bbase = gt4 + ((b * NN) << 2) + k0;

    // software prefetch of the first B-tile
    v2f bnext;
    bnext.x = bbase[(l15 << 2) + 0];
    bnext.y = bbase[(l15 << 2) + 1];

    for (int mt = 0; mt < 64; ++mt) {
        const v2f bcur = bnext;
        if (mt < 63) {
            const float* bp = bbase + (((mt + 1) << 4) + l15) * 4;
            bnext.x = bp[0];
            bnext.y = bp[1];
        }

        v8f acc0 = {}, acc1 = {};
        acc0 = __builtin_amdgcn_wmma_f32_16x16x4_f32(
            false, a0, false, bcur, (short)0, acc0, false, false);
        acc1 = __builtin_amdgcn_wmma_f32_16x16x4_f32(
            false, a1, false, bcur, (short)0, acc1, false, false);

        unsigned int cmin = 0x7F800000u;
        #pragma unroll
        for (int r = 0; r < 8; ++r) {
            const unsigned int u0 =
                __float_as_uint(fmaxf(p2r0[r] + acc0[r], 0.0f));
            const unsigned int u1 =
                __float_as_uint(fmaxf(p2r1[r] + acc1[r], 0.0f));
            rmin0[r] = min(rmin0[r], u0);
            rmin1[r] = min(rmin1[r], u1);
            cmin = min(cmin, min(u0, u1));
        }
        // every lane updates its column; lanes L and L+16 share an address
        // and the L2 atomic unit resolves them -- no shuffle, no branch.
        atomicMin(&dis2u[b * NN + (mt << 4) + l15], cmin);
    }

    // Row mins: 16 lanes per row converge through atomicMin (replaces the
    // ds_bpermute shuffle tree entirely; 16 atomics per lane, once per wave).
    unsigned int* row0 = dis1u + b * NN + n0 + half * 8;
    #pragma unroll
    for (int r = 0; r < 8; ++r) {
        atomicMin(&row0[r],      rmin0[r]);
        atomicMin(&row0[r + 16], rmin1[r]);
    }
}

// ---------------------------------------------------------------------------
// Kernel 3: final scalar reduction. dis1/dis2 hold SQUARED mins -> sqrt here.
// per_point = (1-sym)*l2 + sym*0.5*(sqrt(dis1)+sqrt(dis2));  loss = mean.
// ---------------------------------------------------------------------------
__global__ __launch_bounds__(256) void finalize_kernel(
    const float* __restrict__ l2,
    const float* __restrict__ dis1,
    const float* __restrict__ dis2,
    const float* __restrict__ sym,
    float* __restrict__ out)
{
    __shared__ float red[256];
    float s = 0.0f;
    for (int i = threadIdx.x; i < BB * NN; i += 256) {
        const int   b  = i >> 10;
        const float sf = sym[b];
        const float cd = 0.5f * (__builtin_amdgcn_sqrtf(dis1[i]) +
                                 __builtin_amdgcn_sqrtf(dis2[i]));
        s += (1.0f - sf) * l2[i] + sf * cd;
    }
    red[threadIdx.x] = s;
    __syncthreads();
    for (int o = 128; o > 0; o >>= 1) {
        if (threadIdx.x < o) red[threadIdx.x] += red[threadIdx.x + o];
        __syncthreads();
    }
    if (threadIdx.x == 0)
        out[0] = red[0] * (1.0f / (float)(BB * NN));
}

// ---------------------------------------------------------------------------
extern "C" void kernel_launch(void* const* d_in, const int* in_sizes, int n_in,
                              void* d_out, int out_size, void* d_ws, size_t ws_size,
                              hipStream_t stream)
{
    const float* dR = (const float*)d_in[0];   // delta_rot_pred [B,3,3]
    const float* dT = (const float*)d_in[1];   // delta_trans_pred [B,3]
    const float* Tc = (const float*)d_in[2];   // trans_cur [B,3]
    const float* Rc = (const float*)d_in[3];   // rot_cur [B,3,3]
    const float* Rg = (const float*)d_in[4];   // rot_gt [B,3,3]
    const float* Tg = (const float*)d_in[5];   // trans_gt [B,3]
    const float* P  = (const float*)d_in[6];   // points_tmp [B,N,3]
    const float* sy = (const float*)d_in[7];   // sym_flag [B]
    float* out = (float*)d_out;

    // Workspace layout (floats)
    float* ws    = (float*)d_ws;
    float* pred4 = ws;                         // B*N*4
    float* gt4   = pred4 + BB * NN * 4;        // B*N*4
    float* p2    = gt4   + BB * NN * 4;        // B*N
    float* l2    = p2    + BB * NN;            // B*N
    float* dis1  = l2    + BB * NN;            // B*N
    float* dis2  = dis1  + BB * NN;            // B*N
    unsigned int* dis1u = (unsigned int*)dis1;
    unsigned int* dis2u = (unsigned int*)dis2;

    // 1) fused pose transforms + norms + L2 + min-buffer init
    pose_points_kernel<<<(BB * NN) / 256, 256, 0, stream>>>(
        dR, dT, Tc, Rc, Rg, Tg, P, pred4, gt4, p2, l2, dis1u, dis2u);

    // 2) Chamfer via WMMA: one wave32 per (batch, 32-row strip)
    chamfer_wmma_kernel<<<BB * 32, 32, 0, stream>>>(
        pred4, gt4, p2, dis1u, dis2u);

    // 3) scalar loss
    finalize_kernel<<<1, 256, 0, stream>>>(l2, dis1, dis2, sy, out);
}